// GraphSAGEEncoder_7421703487977
// MI455X (gfx1250) — compile-verified
//
#include <hip/hip_runtime.h>

typedef float v2f __attribute__((ext_vector_type(2)));
typedef float v8f __attribute__((ext_vector_type(8)));

#define N_NODES 50000
#define IN_CH   128
#define HID     128
#define OUT_CH  64

static __device__ __forceinline__ v8f wmma_f32x4(v2f a, v2f b, v8f c) {
  // D = A(16x4 f32) * B(4x16 f32) + C(16x16 f32)
  return __builtin_amdgcn_wmma_f32_16x16x4_f32(false, a, false, b, (short)0, c, false, false);
}

// ---------------------------------------------------------------------------
// Zero a float region
// ---------------------------------------------------------------------------
__global__ void zero_f32(float* __restrict__ p, int n) {
  int i = blockIdx.x * blockDim.x + threadIdx.x;
  if (i < n) p[i] = 0.0f;
}

// ---------------------------------------------------------------------------
// Edge scatter: agg[dst] += feat[src]; cnt[dst] += 1 (once per edge).
// One thread = one (edge, 4-channel group). 32 groups cover 128 channels.
// Gather is a coalesced float4; scatter uses global_atomic_add_f32 (L2 atomics,
// agg resides in L2).
// ---------------------------------------------------------------------------
__global__ void scatter_edges(const float* __restrict__ feat,
                              const long long* __restrict__ edges, // [2, E] int64
                              int E,
                              float* __restrict__ agg,
                              float* __restrict__ cnt,
                              int add_count) {
  long long gid = (long long)blockIdx.x * blockDim.x + threadIdx.x;
  int e   = (int)(gid >> 5);
  int grp = (int)(gid & 31);
  if (e >= E) return;
  int s = (int)edges[e];
  int d = (int)edges[(long long)E + e];

  const float4 v = reinterpret_cast<const float4*>(feat + (size_t)s * IN_CH)[grp];
  float* dp = agg + (size_t)d * IN_CH + grp * 4;
  atomicAdd(dp + 0, v.x);
  atomicAdd(dp + 1, v.y);
  atomicAdd(dp + 2, v.z);
  atomicAdd(dp + 3, v.w);
  if (add_count && grp == 0) atomicAdd(&cnt[d], 1.0f);
}

// ---------------------------------------------------------------------------
// Layer 1 GEMM (fused): h = relu( (agg/max(cnt,1)) @ Wl + b + x @ Wr )
// One wave (32 threads) per 16x16 output tile. K = 128 via 32+32 WMMA ops.
// f32 WMMA operand layout:
//   A 16x4 : lane l holds row M=l&15; VGPR0/1 = K = (l>>4)*2, (l>>4)*2+1
//   B 4x16 : lane l holds col N=l&15; VGPR0/1 = K = (l>>4)*2, (l>>4)*2+1
//   C/D    : VGPR v, lane l -> (M = v + 8*(l>>4), N = l&15)
// ---------------------------------------------------------------------------
__global__ __launch_bounds__(32)
void gemm_layer1(const float* __restrict__ agg, const float* __restrict__ cnt,
                 const float* __restrict__ x,
                 const float* __restrict__ Wl, const float* __restrict__ b,
                 const float* __restrict__ Wr,
                 float* __restrict__ h, int n_nodes) {
  const int mt = blockIdx.x, nt = blockIdx.y;
  const int l = threadIdx.x;
  const int r = l & 15, half = l >> 4;
  int row = mt * 16 + r;
  if (row >= n_nodes) row = n_nodes - 1;          // clamp, EXEC stays full
  const int ncol = nt * 16 + r;
  const int kofs = half * 2;

  const float inv = 1.0f / fmaxf(cnt[row], 1.0f);
  const float* __restrict__ aA = agg + (size_t)row * HID;
  const float* __restrict__ xA = x   + (size_t)row * IN_CH;

  v8f c = {0.f, 0.f, 0.f, 0.f, 0.f, 0.f, 0.f, 0.f};

#pragma unroll 4
  for (int k0 = 0; k0 < HID; k0 += 4) {
    v2f a;
    a.x = aA[k0 + kofs + 0] * inv;
    a.y = aA[k0 + kofs + 1] * inv;
    v2f bv;
    bv.x = Wl[(size_t)(k0 + kofs + 0) * HID + ncol];
    bv.y = Wl[(size_t)(k0 + kofs + 1) * HID + ncol];
    c = wmma_f32x4(a, bv, c);
  }
#pragma unroll 4
  for (int k0 = 0; k0 < IN_CH; k0 += 4) {
    v2f a = *reinterpret_cast<const v2f*>(xA + k0 + kofs);
    v2f bv;
    bv.x = Wr[(size_t)(k0 + kofs + 0) * HID + ncol];
    bv.y = Wr[(size_t)(k0 + kofs + 1) * HID + ncol];
    c = wmma_f32x4(a, bv, c);
  }

  const float bias = b[ncol];
#pragma unroll
  for (int v = 0; v < 8; ++v) {
    int orow = mt * 16 + v + 8 * half;
    if (orow < n_nodes)
      h[(size_t)orow * HID + ncol] = fmaxf(c[v] + bias, 0.0f);
  }
}

// ---------------------------------------------------------------------------
// Layer 2 GEMM (fused, mu + logstd share the A matrices):
//   mu = mean_h @ Wmu_l + bmu + h @ Wmu_r
//   ls = mean_h @ Wls_l + bls + h @ Wls_r
// ---------------------------------------------------------------------------
__global__ __launch_bounds__(32)
void gemm_layer2(const float* __restrict__ aggh, const float* __restrict__ cnt,
                 const float* __restrict__ h,
                 const float* __restrict__ Wmu_l, const float* __restrict__ bmu,
                 const float* __restrict__ Wmu_r,
                 const float* __restrict__ Wls_l, const float* __restrict__ bls,
                 const float* __restrict__ Wls_r,
                 float* __restrict__ out_mu, float* __restrict__ out_ls,
                 int n_nodes) {
  const int mt = blockIdx.x, nt = blockIdx.y;
  const int l = threadIdx.x;
  const int r = l & 15, half = l >> 4;
  int row = mt * 16 + r;
  if (row >= n_nodes) row = n_nodes - 1;
  const int ncol = nt * 16 + r;
  const int kofs = half * 2;

  const float inv = 1.0f / fmaxf(cnt[row], 1.0f);
  const float* __restrict__ aA = aggh + (size_t)row * HID;
  const float* __restrict__ hA = h    + (size_t)row * HID;

  v8f cmu = {0.f, 0.f, 0.f, 0.f, 0.f, 0.f, 0.f, 0.f};
  v8f cls = {0.f, 0.f, 0.f, 0.f, 0.f, 0.f, 0.f, 0.f};

#pragma unroll 4
  for (int k0 = 0; k0 < HID; k0 += 4) {
    v2f a;
    a.x = aA[k0 + kofs + 0] * inv;
    a.y = aA[k0 + kofs + 1] * inv;
    v2f bmu_v, bls_v;
    bmu_v.x = Wmu_l[(size_t)(k0 + kofs + 0) * OUT_CH + ncol];
    bmu_v.y = Wmu_l[(size_t)(k0 + kofs + 1) * OUT_CH + ncol];
    bls_v.x = Wls_l[(size_t)(k0 + kofs + 0) * OUT_CH + ncol];
    bls_v.y = Wls_l[(size_t)(k0 + kofs + 1) * OUT_CH + ncol];
    cmu = wmma_f32x4(a, bmu_v, cmu);
    cls = wmma_f32x4(a, bls_v, cls);
  }
#pragma unroll 4
  for (int k0 = 0; k0 < HID; k0 += 4) {
    v2f a = *reinterpret_cast<const v2f*>(hA + k0 + kofs);
    v2f bmu_v, bls_v;
    bmu_v.x = Wmu_r[(size_t)(k0 + kofs + 0) * OUT_CH + ncol];
    bmu_v.y = Wmu_r[(size_t)(k0 + kofs + 1) * OUT_CH + ncol];
    bls_v.x = Wls_r[(size_t)(k0 + kofs + 0) * OUT_CH + ncol];
    bls_v.y = Wls_r[(size_t)(k0 + kofs + 1) * OUT_CH + ncol];
    cmu = wmma_f32x4(a, bmu_v, cmu);
    cls = wmma_f32x4(a, bls_v, cls);
  }

  const float biasmu = bmu[ncol];
  const float biasls = bls[ncol];
#pragma unroll
  for (int v = 0; v < 8; ++v) {
    int orow = mt * 16 + v + 8 * half;
    if (orow < n_nodes) {
      out_mu[(size_t)orow * OUT_CH + ncol] = cmu[v] + biasmu;
      out_ls[(size_t)orow * OUT_CH + ncol] = cls[v] + biasls;
    }
  }
}

// ---------------------------------------------------------------------------
// Host launch
// ---------------------------------------------------------------------------
extern "C" void kernel_launch(void* const* d_in, const int* in_sizes, int n_in,
                              void* d_out, int out_size, void* d_ws, size_t ws_size,
                              hipStream_t stream) {
  const float*     x     = (const float*)d_in[0];
  const long long* edges = (const long long*)d_in[1];
  const float*     W1_l  = (const float*)d_in[2];
  const float*     b1    = (const float*)d_in[3];
  const float*     W1_r  = (const float*)d_in[4];
  const float*     Wmu_l = (const float*)d_in[5];
  const float*     bmu   = (const float*)d_in[6];
  const float*     Wmu_r = (const float*)d_in[7];
  const float*     Wls_l = (const float*)d_in[8];
  const float*     bls   = (const float*)d_in[9];
  const float*     Wls_r = (const float*)d_in[10];

  const int N = in_sizes[0] / IN_CH;   // 50000
  const int E = in_sizes[1] / 2;       // 800000

  // Workspace layout (floats): [cnt: N][agg: N*HID][h: N*HID]
  float* cnt = (float*)d_ws;
  float* agg = cnt + N;
  float* h   = agg + (size_t)N * HID;

  float* out_mu = (float*)d_out;
  float* out_ls = out_mu + (size_t)N * OUT_CH;

  const int mtiles = (N + 15) / 16;

  // 1) zero cnt + agg (contiguous)
  {
    int n = N * (1 + HID);
    zero_f32<<<(n + 255) / 256, 256, 0, stream>>>(cnt, n);
  }
  // 2) aggregate x over edges (+degree counts)
  {
    long long threads = (long long)E * 32;
    int blocks = (int)((threads + 255) / 256);
    scatter_edges<<<blocks, 256, 0, stream>>>(x, edges, E, agg, cnt, 1);
  }
  // 3) layer-1 fused GEMM -> h
  {
    dim3 grid(mtiles, HID / 16);
    gemm_layer1<<<grid, 32, 0, stream>>>(agg, cnt, x, W1_l, b1, W1_r, h, N);
  }
  // 4) zero agg (cnt is reused unchanged)
  {
    int n = N * HID;
    zero_f32<<<(n + 255) / 256, 256, 0, stream>>>(agg, n);
  }
  // 5) aggregate h over edges
  {
    long long threads = (long long)E * 32;
    int blocks = (int)((threads + 255) / 256);
    scatter_edges<<<blocks, 256, 0, stream>>>(h, edges, E, agg, nullptr, 0);
  }
  // 6) layer-2 fused GEMM -> mu, logstd
  {
    dim3 grid(mtiles, OUT_CH / 16);
    gemm_layer2<<<grid, 32, 0, stream>>>(agg, cnt, h, Wmu_l, bmu, Wmu_r,
                                         Wls_l, bls, Wls_r, out_mu, out_ls, N);
  }
}